// hierarchical_cluster_assignment_52544629899790
// MI455X (gfx1250) — compile-verified
//
#include <hip/hip_runtime.h>

// Hierarchical greedy merge clustering, one workgroup per (l,n) problem.
// Everything lives in LDS (needs CDNA5's 320KB WGP LDS). Initial Gram via
// V_WMMA_F32_16X16X4_F32; thereafter O(Q*C) incremental distance updates.
// Argmin + norm reductions use wave32 shuffle butterflies (critical path).

typedef __attribute__((ext_vector_type(2))) float v2f;
typedef __attribute__((ext_vector_type(8))) float v8f;

#define TQ   64    // initial centers per problem
#define TC   256   // channels
#define CSTR 260   // LDS float stride for center rows: 1040B rows -> 16B aligned, 260%64==4 -> conflict-free column walks
#define DSTR 65    // LDS float stride for the d^2 matrix
#define NT   128   // 4 waves of 32
#define NW   (NT / 32)

__global__ __launch_bounds__(NT, 1)
void hac_merge_kernel(const float* __restrict__ in, float* __restrict__ out)
{
    __shared__ float sC[TQ * CSTR];   // centers, 66,560 B
    __shared__ float sD[TQ * DSTR];   // squared distances, 16,640 B
    __shared__ float sSq[TQ];         // squared norms
    __shared__ float sRedV[NW];
    __shared__ int   sRedI[NW];
    __shared__ int   sXY[2];

    const int tid  = threadIdx.x;
    const int blk  = blockIdx.x;
    const int lane = tid & 31;
    const int wave = tid >> 5;

    // ---------------- load this problem's 64x256 tile ----------------
    const float4* gin = (const float4*)(in + (size_t)blk * (TQ * TC));
    for (int i = tid; i < TQ * TC / 4; i += NT) {
        float4 v = gin[i];
        int q  = i >> 6;          // 64 float4 per row
        int c4 = (i & 63) << 2;
        *(float4*)&sC[q * CSTR + c4] = v;
    }
    __syncthreads();

    // ---------------- squared norms ----------------
    if (tid < TQ) {
        const float* row = &sC[tid * CSTR];
        float s = 0.f;
        for (int c = 0; c < TC; ++c) s = fmaf(row[c], row[c], s);
        sSq[tid] = s;
    }

    // ---------------- initial 64x64 Gram via WMMA f32 16x16x4 ----------------
    {
        const int m  = lane & 15;   // M (A) / N (B) index
        const int kp = lane >> 4;   // K pair select: lanes 0-15 -> K=0,1 ; 16-31 -> K=2,3
        for (int t = wave; t < 16; t += NW) {         // 4 tiles per wave
            const int ti = t >> 2, tj = t & 3;
            const float* arow = &sC[(ti * 16 + m) * CSTR + 2 * kp];
            const float* brow = &sC[(tj * 16 + m) * CSTR + 2 * kp];
            v8f acc = {};
            for (int kb = 0; kb < TC; kb += 4) {
                v2f a = *(const v2f*)(arow + kb);     // A: 16x4 f32 fragment
                v2f b = *(const v2f*)(brow + kb);     // B: 4x16 f32 fragment (Gram is symmetric)
                acc = __builtin_amdgcn_wmma_f32_16x16x4_f32(
                        false, a, false, b, (short)0, acc, false, false);
            }
            // C/D layout: VGPR r, lanes<16 -> M=r, lanes>=16 -> M=r+8
            const int rbase = ti * 16 + ((lane >> 4) << 3);
            const int col   = tj * 16 + m;
#pragma unroll
            for (int r = 0; r < 8; ++r)
                sD[(rbase + r) * DSTR + col] = acc[r];
        }
    }
    __syncthreads();

    // ---------------- gram -> squared distance ----------------
    for (int i = tid; i < TQ * TQ; i += NT) {
        int q = i >> 6, p = i & 63;
        float g = sD[q * DSTR + p];
        sD[q * DSTR + p] = sSq[q] + sSq[p] - 2.f * g;
    }
    __syncthreads();

    // ---------------- 62 greedy merge rounds ----------------
    for (int Q = TQ; Q >= 3; --Q) {
        // Argmin over off-diagonal clamped d^2. The reference's flattened
        // q*Q+p tie-break order is lexicographic (q,p), so scanning the fixed
        // 64-wide layout with tie-break key q*64+p picks the same pair
        // (and symmetry + first-index tie-break guarantees x < y).
        float bv = 3.4e38f;
        int   bi = 0x7fffffff;
        for (int i = tid; i < TQ * TQ; i += NT) {
            int q = i >> 6, p = i & 63;
            if (q >= Q || p >= Q || q == p) continue;
            float v = sD[q * DSTR + p];
            v = v > 0.f ? v : 0.f;                    // matches sqrt(max(d2,0)) ordering
            if (v < bv || (v == bv && i < bi)) { bv = v; bi = i; }
        }
        // wave32 butterfly reduce (value, index) with min-index tie-break
#pragma unroll
        for (int off = 16; off > 0; off >>= 1) {
            float ov = __shfl_xor(bv, off, 32);
            int   oi = __shfl_xor(bi, off, 32);
            if (ov < bv || (ov == bv && oi < bi)) { bv = ov; bi = oi; }
        }
        if (lane == 0) { sRedV[wave] = bv; sRedI[wave] = bi; }
        __syncthreads();
        if (tid == 0) {
            float mv = sRedV[0]; int mi = sRedI[0];
#pragma unroll
            for (int w = 1; w < NW; ++w) {
                float v = sRedV[w]; int ii = sRedI[w];
                if (v < mv || (v == mv && ii < mi)) { mv = v; mi = ii; }
            }
            sXY[0] = mi >> 6;
            sXY[1] = mi & 63;
        }
        __syncthreads();
        const int x = sXY[0], y = sXY[1];   // x < y

        // centers[x] = 0.5*(cx+cy); centers[y] = centers[Q-1]
        for (int c = tid; c < TC; c += NT) {
            float cx = sC[x * CSTR + c], cy = sC[y * CSTR + c];
            sC[x * CSTR + c] = 0.5f * (cx + cy);
            sC[y * CSTR + c] = sC[(Q - 1) * CSTR + c];
        }
        __syncthreads();

        // norms: x recomputed from merged center (wave 0, deterministic tree);
        // y inherits old last-row norm (exact copy keeps d^2 symmetric)
        if (wave == 0) {
            const float* row = &sC[x * CSTR];
            float s = 0.f;
#pragma unroll
            for (int c = 0; c < TC / 32; ++c) {
                float v = row[lane + c * 32];
                s = fmaf(v, v, s);
            }
#pragma unroll
            for (int off = 16; off > 0; off >>= 1)
                s += __shfl_xor(s, off, 32);
            if (lane == 0) sSq[x] = s;
        }
        if (tid == 32) sSq[y] = sSq[Q - 1];  // x != Q-1; y==Q-1 is a no-op
        __syncthreads();

        // refresh only rows/cols x and y vs the surviving Q-1 centers
        const int Qp = Q - 1;                // 2*Qp <= 126 < NT
        if (tid < 2 * Qp) {
            int r = (tid < Qp) ? x : y;
            int p = (tid < Qp) ? tid : tid - Qp;
            const float* ra = &sC[r * CSTR];
            const float* rb = &sC[p * CSTR];
            float s = 0.f;
            for (int c = 0; c < TC; ++c) s = fmaf(ra[c], rb[c], s);
            float v = sSq[r] + sSq[p] - 2.f * s;
            sD[r * DSTR + p] = v;            // (x,y)/(y,x) pair writes identical values
            sD[p * DSTR + r] = v;
        }
        __syncthreads();
    }

    // ---------------- emit the final 2 centers ----------------
    for (int i = tid; i < 2 * TC; i += NT)
        out[(size_t)blk * (2 * TC) + i] = sC[(i >> 8) * CSTR + (i & 255)];
}

extern "C" void kernel_launch(void* const* d_in, const int* in_sizes, int n_in,
                              void* d_out, int out_size, void* d_ws, size_t ws_size,
                              hipStream_t stream)
{
    const float* in  = (const float*)d_in[0];
    float*       out = (float*)d_out;
    const int batches = in_sizes[0] / (TQ * TC);   // L*N = 384
    hac_merge_kernel<<<dim3(batches), dim3(NT), 0, stream>>>(in, out);
}